// AttentionM_20761871908922
// MI455X (gfx1250) — compile-verified
//
#include <hip/hip_runtime.h>
#include <math.h>

typedef __attribute__((ext_vector_type(2))) float v2f;
typedef __attribute__((ext_vector_type(8))) float v8f;

#define BB 32
#define TT 4096
#define DD 512

// ---------------------------------------------------------------------------
// Pass 1: et[b*T + t] = tanh( dot(x[b,t,:], W) + bias[t] )
// One wave handles 16 consecutive (b,t) rows via V_WMMA_F32_16X16X4_F32.
//   A (16x4 f32): lane l<16  -> a = { x[row=l][k+0], x[row=l][k+1] }
//                 lane l>=16 -> a = { x[row=l-16][k+2], x[row=l-16][k+3] }
//   B (4x16 f32): W[k..k+3] broadcast across all 16 columns (same slot map).
// All 16 result columns are identical; extract column N=0:
//   lane 0  holds M=0..7  in c[0..7], lane 16 holds M=8..15 in c[0..7].
// ---------------------------------------------------------------------------
__global__ __launch_bounds__(256)
void et_gemv_wmma_kernel(const float* __restrict__ x,
                         const float* __restrict__ W,
                         const float* __restrict__ bias,
                         float* __restrict__ et) {
    const int lane = threadIdx.x & 31;
    const int wave = threadIdx.x >> 5;
    const long tile = (long)blockIdx.x * 8 + wave;   // 8192 tiles of 16 rows
    const long r0 = tile * 16;                       // flat row index (b*T + t)

    const int m    = lane & 15;
    const int half = lane >> 4;

    const float* xrow = x + (r0 + m) * DD;

    v8f c = {};
    #pragma unroll 4
    for (int k = 0; k < DD; k += 4) {
        const int kk = k + half * 2;
        v2f a  = *(const v2f*)(xrow + kk);   // global_load_b64
        v2f bf = *(const v2f*)(W + kk);      // W broadcast (hits cache)
        c = __builtin_amdgcn_wmma_f32_16x16x4_f32(
                /*neg_a=*/false, a, /*neg_b=*/false, bf,
                /*c_mod=*/(short)0, c, /*reuse_a=*/false, /*reuse_b=*/false);
    }

    if ((lane & 15) == 0) {                  // lanes 0 and 16
        union { v8f v; float f[8]; } u; u.v = c;
        const int mbase = half * 8;
        #pragma unroll
        for (int i = 0; i < 8; ++i) {
            const long r = r0 + mbase + i;
            const int  t = (int)(r & (TT - 1));
            et[r] = tanhf(u.f[i] + bias[t]);
        }
    }
}

// ---------------------------------------------------------------------------
// Pass 2: in-place softmax over T for each batch row. One block per b.
// ---------------------------------------------------------------------------
__global__ __launch_bounds__(256)
void softmax_kernel(float* __restrict__ et) {
    __shared__ float red[256];
    const int b   = blockIdx.x;
    const int tid = threadIdx.x;
    float* row = et + (long)b * TT;

    float mx = -INFINITY;
    for (int i = tid; i < TT; i += 256) mx = fmaxf(mx, row[i]);
    red[tid] = mx; __syncthreads();
    for (int s = 128; s > 0; s >>= 1) {
        if (tid < s) red[tid] = fmaxf(red[tid], red[tid + s]);
        __syncthreads();
    }
    mx = red[0]; __syncthreads();

    float sum = 0.0f;
    for (int i = tid; i < TT; i += 256) {
        const float e = __expf(row[i] - mx);
        row[i] = e;
        sum += e;
    }
    red[tid] = sum; __syncthreads();
    for (int s = 128; s > 0; s >>= 1) {
        if (tid < s) red[tid] += red[tid + s];
        __syncthreads();
    }
    const float inv = 1.0f / red[0];
    for (int i = tid; i < TT; i += 256) row[i] *= inv;
}

// ---------------------------------------------------------------------------
// Pass 3: partial pooling. Block = (t-chunk of 128, batch b).
// Thread owns d = 2*tid, 2*tid+1 -> one coalesced b64 load per row.
// ---------------------------------------------------------------------------
__global__ __launch_bounds__(256)
void pool_partial_kernel(const float* __restrict__ x,
                         const float* __restrict__ at,
                         float* __restrict__ partial) {
    __shared__ float aw[128];
    const int chunk = blockIdx.x;   // 0..31
    const int b     = blockIdx.y;   // 0..31
    const int tid   = threadIdx.x;  // 0..255

    if (tid < 128) aw[tid] = at[(long)b * TT + chunk * 128 + tid];
    __syncthreads();

    const float* xb = x + ((long)b * TT + (long)chunk * 128) * DD;

    float ax = 0.0f, ay = 0.0f;
    for (int t = 0; t < 128; ++t) {
        const float a = aw[t];
        v2f xv = *(const v2f*)(xb + (long)t * DD + 2 * tid);
        ax += a * xv.x;
        ay += a * xv.y;
    }
    v2f acc = { ax, ay };
    *(v2f*)(partial + ((long)(b * 32 + chunk)) * DD + 2 * tid) = acc;
}

// ---------------------------------------------------------------------------
// Pass 4: reduce the 32 chunk-partials -> out (B x D), fully overwritten.
// ---------------------------------------------------------------------------
__global__ __launch_bounds__(256)
void pool_reduce_kernel(const float* __restrict__ partial,
                        float* __restrict__ out) {
    const int b   = blockIdx.x;     // 0..31
    const int tid = threadIdx.x;    // 0..255
    float ax = 0.0f, ay = 0.0f;
    #pragma unroll 4
    for (int chunk = 0; chunk < 32; ++chunk) {
        v2f p = *(const v2f*)(partial + ((long)(b * 32 + chunk)) * DD + 2 * tid);
        ax += p.x;
        ay += p.y;
    }
    v2f acc = { ax, ay };
    *(v2f*)(out + (long)b * DD + 2 * tid) = acc;
}

extern "C" void kernel_launch(void* const* d_in, const int* in_sizes, int n_in,
                              void* d_out, int out_size, void* d_ws, size_t ws_size,
                              hipStream_t stream) {
    const float* x    = (const float*)d_in[0];  // (B, T, D) fp32
    const float* W    = (const float*)d_in[1];  // (D, 1)    fp32
    const float* bias = (const float*)d_in[2];  // (T, 1)    fp32
    float* out = (float*)d_out;                 // (B, D)    fp32

    float* wsf     = (float*)d_ws;
    float* et      = wsf;                       // B*T floats      (et -> at in place)
    float* partial = wsf + (long)BB * TT;       // B*32*D floats

    // Pass 1: 8192 row-tiles of 16, 8 waves/block -> 1024 blocks.
    et_gemv_wmma_kernel<<<dim3(1024), dim3(256), 0, stream>>>(x, W, bias, et);
    // Pass 2: softmax per batch.
    softmax_kernel<<<dim3(BB), dim3(256), 0, stream>>>(et);
    // Pass 3: partial weighted pooling, 32 chunks x 32 batches.
    pool_partial_kernel<<<dim3(32, BB), dim3(256), 0, stream>>>(x, et, partial);
    // Pass 4: final reduction into d_out.
    pool_reduce_kernel<<<dim3(BB), dim3(256), 0, stream>>>(partial, out);
}